// SinkhornScorer_73418170958112
// MI455X (gfx1250) — compile-verified
//
#include <hip/hip_runtime.h>
#include <stdint.h>

typedef __attribute__((ext_vector_type(2))) float v2f;
typedef __attribute__((ext_vector_type(8))) float v8f;
typedef __attribute__((ext_vector_type(4))) unsigned int v4u;
typedef __attribute__((ext_vector_type(8))) int v8i;
typedef __attribute__((ext_vector_type(4))) int v4i;

#define MM 256
#define PITCH 257
#define NORMC (-6.2383246250395075f)   /* -log(512) */
#define LOG256 (5.545177444479562f)    /*  log(256) */

__global__ __launch_bounds__(1024)
void sinkhorn_kernel(const float* __restrict__ scores,
                     const float* __restrict__ alpha_p,
                     const int* __restrict__ iters_p,
                     float* __restrict__ out)
{
    // K = exp(couplings), row pitch 257 dwords (257 % 64 == 1 -> conflict-free
    // for both row-major and column-major access patterns).
    __shared__ float Kmat[PITCH * PITCH];
    __shared__ float w_arr[PITCH];   // exp(v)
    __shared__ float t_arr[PITCH];   // exp(u)
    __shared__ float u_arr[PITCH];
    __shared__ float v_arr[PITCH];

    const int tid  = threadIdx.x;
    const int lane = tid & 31;
    const int wave = tid >> 5;
    const int b    = blockIdx.x;

    const float alphav = alpha_p[0];
    const int   iters  = iters_p[0];
    const float ea     = __expf(alphav);
    const size_t bscore = (size_t)b * (MM * MM);
    const size_t bout   = (size_t)b * (PITCH * PITCH);

    // ---- Phase 0: TDM async-DMA scores[b] (256x256 f32) -> LDS with
    // 1-dword pad per 256 dwords, i.e. the 257-dword pitch we want. ----
    if (wave == 0) {
        uint32_t ldsa = (uint32_t)(uintptr_t)(&Kmat[0]);
        uint64_t ga   = (uint64_t)(uintptr_t)(scores + bscore);
        v4u g0 = { 1u,                               // count=1, user mode
                   ldsa,                             // lds_addr
                   (uint32_t)ga,                     // global_addr[31:0]
                   (uint32_t)((ga >> 32) & 0x1FFFFFFull) | (2u << 30) }; // [56:32] | type=2
        v8i g1 = { (int)((2u << 16) | (1u << 20) | (7u << 22)),
                   // data_size=4B, pad_enable, pad_interval=256dw, pad_amount=1dw
                   (int)(256u << 16),                // tensor_dim0 = 256 (low 16 in [31:16])
                   (int)(256u << 16),                // dim0_hi=0 | tensor_dim1 = 256
                   (int)(256u << 16),                // dim1_hi=0 | tile_dim0 = 256
                   (int)256,                         // tile_dim1=256, tile_dim2=0
                   (int)256,                         // tensor_dim0_stride = 256
                   0, 0 };
        v4i g2 = { 1, 0, 0, 0 };                     // tensor_dim2=1 (unused, 2D tile)
        v4i g3 = { 0, 0, 0, 0 };
        v8i gx = { 0, 0, 0, 0, 0, 0, 0, 0 };         // aux slot (clang-23 6-arg form)
        __builtin_amdgcn_tensor_load_to_lds(g0, g1, g2, g3, gx, 0);
    }

    // Concurrently with the DMA: dustbin row/col (disjoint LDS) + w=exp(v0)=1.
    for (int i = tid; i < PITCH; i += 1024) {
        Kmat[MM * PITCH + i] = ea;                    // bottom row incl. corner
        if (i < MM) Kmat[i * PITCH + MM] = ea;        // right column
        w_arr[i] = 1.0f;
    }

    if (wave == 0) __builtin_amdgcn_s_wait_tensorcnt(0);
    __syncthreads();

    // exp() in place over the 256x256 interior (conflict-free, coalesced).
    for (int idx = tid; idx < MM * MM; idx += 1024) {
        int m = idx >> 8, n = idx & 255;
        Kmat[m * PITCH + n] = __expf(Kmat[m * PITCH + n]);
    }
    __syncthreads();

    // ---- Sinkhorn iterations in exp-domain: LSE passes become GEMVs. ----
    for (int it = 0; it < iters; ++it) {
        // pass 1: u = log_mu - log(K @ w)   (one wave per row)
        float wreg[9];
#pragma unroll
        for (int j = 0; j < 9; ++j) {
            int n = lane + 32 * j;
            wreg[j] = (n < PITCH) ? w_arr[n] : 0.0f;
        }
        for (int m = wave; m < PITCH; m += 32) {
            const float* row = &Kmat[m * PITCH];
            float acc = 0.0f;
#pragma unroll
            for (int j = 0; j < 9; ++j) {
                int n = lane + 32 * j;
                if (n < PITCH) acc += row[n] * wreg[j];
            }
#pragma unroll
            for (int off = 16; off > 0; off >>= 1) acc += __shfl_xor(acc, off, 32);
            if (lane == 0) {
                float lmu = (m < MM) ? NORMC : (LOG256 + NORMC);
                float um  = lmu - __logf(acc);
                u_arr[m] = um;
                t_arr[m] = __expf(um);
            }
        }
        __syncthreads();

        // pass 2: v = log_nu - log(K^T @ t)   (one wave per column)
        float treg[9];
#pragma unroll
        for (int j = 0; j < 9; ++j) {
            int m = lane + 32 * j;
            treg[j] = (m < PITCH) ? t_arr[m] : 0.0f;
        }
        for (int n = wave; n < PITCH; n += 32) {
            float acc = 0.0f;
#pragma unroll
            for (int j = 0; j < 9; ++j) {
                int m = lane + 32 * j;
                if (m < PITCH) acc += Kmat[m * PITCH + n] * treg[j];
            }
#pragma unroll
            for (int off = 16; off > 0; off >>= 1) acc += __shfl_xor(acc, off, 32);
            if (lane == 0) {
                float lnu = (n < MM) ? NORMC : (LOG256 + NORMC);
                float vn  = lnu - __logf(acc);
                v_arr[n] = vn;
                w_arr[n] = __expf(vn);
            }
        }
        __syncthreads();
    }

    // ---- Epilogue: Z = c + u + v - norm ----
    // dustbin row & column (scalar)
    for (int i = tid; i < PITCH; i += 1024) {
        out[bout + MM * PITCH + i] = alphav + u_arr[MM] + v_arr[i] - NORMC;
        if (i < MM) out[bout + i * PITCH + MM] = alphav + u_arr[i] + v_arr[MM] - NORMC;
    }
    // 256x256 interior as WMMA rank-2 update: D = A(16x4) * B(4x16) + C
    //   A rows = [u[m]-norm, 1, 0, 0],  B = [ones; v[n]; 0; 0],  C = scores tile
    const float* sb = scores + bscore;
    const int li = lane & 15;   // N within tile (and M index for A/B builders)
    const int hi = lane >> 4;   // 0: lanes 0-15 (k=0,1 / M=r), 1: lanes 16-31 (k=2,3 / M=r+8)
    for (int t = wave; t < 256; t += 32) {
        int tm = (t >> 4) * 16;
        int tn = (t & 15) * 16;
        float uu = u_arr[tm + li] - NORMC;
        float vv = v_arr[tn + li];
        v2f a, bm;
        a.x  = hi ? 0.0f : uu;   a.y  = hi ? 0.0f : 1.0f;   // A: k0=u-norm, k1=1
        bm.x = hi ? 0.0f : 1.0f; bm.y = hi ? 0.0f : vv;     // B: k0=ones,   k1=v
        v8f c;
#pragma unroll
        for (int r = 0; r < 8; ++r)
            c[r] = sb[(tm + r + hi * 8) * MM + tn + li];
        v8f d = __builtin_amdgcn_wmma_f32_16x16x4_f32(
            /*neg_a=*/false, a, /*neg_b=*/false, bm,
            /*c_mod=*/(short)0, c, /*reuse_a=*/false, /*reuse_b=*/false);
#pragma unroll
        for (int r = 0; r < 8; ++r)
            out[bout + (size_t)(tm + r + hi * 8) * PITCH + tn + li] = d[r];
    }
}

extern "C" void kernel_launch(void* const* d_in, const int* in_sizes, int n_in,
                              void* d_out, int out_size, void* d_ws, size_t ws_size,
                              hipStream_t stream) {
    const float* scores = (const float*)d_in[0];
    const float* alpha  = (const float*)d_in[1];
    const int*   iters  = (const int*)d_in[2];
    float* out = (float*)d_out;
    (void)in_sizes; (void)n_in; (void)out_size; (void)d_ws; (void)ws_size;
    sinkhorn_kernel<<<512, 1024, 0, stream>>>(scores, alpha, iters, out);
}